// Binarization_69930657513784
// MI455X (gfx1250) — compile-verified
//
#include <hip/hip_runtime.h>
#include <stdint.h>

// Reference: adaptive_high_order_residual_v2, ORDER=2, x:(4096,11008) f32, mask bool.
// Bandwidth-bound (405 MB traffic, ~0.7 GFLOP): read x once via the CDNA5 Tensor
// Data Mover into LDS, keep the row in registers for all passes, write out once.

#define COLS 11008
#define TPB  256
#define EPT  43                       // 256 * 43 = 11008
#define SQRT_2_OVER_PI 0.79788456080286535588f

typedef unsigned int u32;
typedef unsigned int v4u __attribute__((ext_vector_type(4)));
typedef int          v8i __attribute__((ext_vector_type(8)));
typedef int          v4i __attribute__((ext_vector_type(4)));

__device__ __forceinline__ float wave_sum(float v) {
#pragma unroll
  for (int o = 16; o > 0; o >>= 1) v += __shfl_xor(v, o, 32);   // wave32
  return v;
}

// Reduce three values across a 256-thread (8-wave) block. buf: 24 floats in LDS.
__device__ __forceinline__ void block_reduce3(float& a, float& b, float& c, float* buf) {
  const int lane = threadIdx.x & 31;
  const int wid  = threadIdx.x >> 5;
  a = wave_sum(a); b = wave_sum(b); c = wave_sum(c);
  if (lane == 0) { buf[wid * 3 + 0] = a; buf[wid * 3 + 1] = b; buf[wid * 3 + 2] = c; }
  __syncthreads();
  float ra = 0.f, rb = 0.f, rc = 0.f;
  if (threadIdx.x < 8) {
    ra = buf[threadIdx.x * 3 + 0];
    rb = buf[threadIdx.x * 3 + 1];
    rc = buf[threadIdx.x * 3 + 2];
  }
  if (wid == 0) {
    ra = wave_sum(ra); rb = wave_sum(rb); rc = wave_sum(rc);
    if (lane == 0) { buf[0] = ra; buf[1] = rb; buf[2] = rc; }
  }
  __syncthreads();
  a = buf[0]; b = buf[1]; c = buf[2];
  __syncthreads();  // buf is reused by the next reduction
}

#if defined(__gfx1250__) && __has_builtin(__builtin_amdgcn_tensor_load_to_lds)
// Build a D# for a 1-row (1 x dim) tile, per CDNA5 ISA.
// data_size_code: 0=1B, 1=2B, 2=4B. dim/stride are in data_size units.
__device__ __forceinline__ void make_row_desc(uint64_t gaddr, u32 lds_off, u32 data_size_code,
                                              u32 dim, v4u& g0, v8i& g1) {
  g0.x = 1u;                                                // count=1, user, no gather
  g0.y = lds_off;                                           // lds_addr [63:32]
  g0.z = (u32)gaddr;                                        // global_addr[31:0]
  g0.w = (u32)((gaddr >> 32) & 0x01FFFFFFu) | (2u << 30);   // global_addr[56:32] | type=2
  g1[0] = (int)(data_size_code << 16);                      // wg_mask=0 | data_size
  g1[1] = (int)((dim & 0xFFFFu) << 16);                     // tensor_dim0 lo16 @bits[63:48]
  g1[2] = (int)((dim >> 16) | (1u << 16));                  // tensor_dim0 hi16 | tensor_dim1=1 lo16
  g1[3] = (int)(dim << 16);                                 // tensor_dim1 hi16 | tile_dim0=dim
  g1[4] = (int)1u;                                          // tile_dim1=1, tile_dim2=0
  g1[5] = (int)dim;                                         // tensor_dim0_stride lo32
  g1[6] = 0;                                                // stride hi16 | dim1_stride lo16
  g1[7] = 0;
}
#endif

__global__ __launch_bounds__(TPB) void binarize2_rowkernel(
    const float* __restrict__ x, const unsigned char* __restrict__ mask,
    float* __restrict__ out) {
  __shared__ alignas(16) float          sx[COLS];   // 44 KB
  __shared__ alignas(16) unsigned char  sm[COLS];   // 11 KB
  __shared__ float red[24];

  const int row = blockIdx.x;
  const int tid = threadIdx.x;
  const float* __restrict__        xr = x    + (size_t)row * COLS;
  const unsigned char* __restrict__ mr = mask + (size_t)row * COLS;

#if defined(__gfx1250__) && __has_builtin(__builtin_amdgcn_tensor_load_to_lds)
  if (tid == 0) {  // one TDM issue per block; EXEC is ignored by tensor ops
    v4u g0; v8i g1;
    v4i gz = {0, 0, 0, 0};
    v8i zz = {0, 0, 0, 0, 0, 0, 0, 0};
    make_row_desc((uint64_t)(uintptr_t)xr, (u32)(uintptr_t)&sx[0], 2u, (u32)COLS, g0, g1);
    __builtin_amdgcn_tensor_load_to_lds(g0, g1, gz, gz, zz, 0);
    make_row_desc((uint64_t)(uintptr_t)mr, (u32)(uintptr_t)&sm[0], 0u, (u32)COLS, g0, g1);
    __builtin_amdgcn_tensor_load_to_lds(g0, g1, gz, gz, zz, 0);
    __builtin_amdgcn_s_wait_tensorcnt(0);
  }
#else
  for (int i = tid; i < COLS; i += TPB) sx[i] = xr[i];
  for (int i = tid; i < COLS; i += TPB) sm[i] = mr[i];
#endif
  __syncthreads();

  // ---- pull row into registers once; pack mask into a 64-bit bitmask ----
  float v[EPT];
  uint64_t mb = 0ull;
  float cnt = 0.f, sum = 0.f, zero = 0.f;
#pragma unroll
  for (int i = 0; i < EPT; ++i) {
    const int idx = tid + i * TPB;
    const bool m  = sm[idx] != 0;
    const float xm = m ? sx[idx] : 0.f;
    v[i] = xm;
    mb  |= m ? (1ull << i) : 0ull;
    cnt += m ? 1.f : 0.f;
    sum += xm;
  }
  block_reduce3(cnt, sum, zero, red);
  const float inv_cnt = (cnt > 0.f) ? (1.f / fmaxf(cnt, 1.f)) : 0.f;
  const float mean1   = sum * inv_cnt;

  // ---- order 1: center, accumulate sum(c), sum(c^2), sum(sign(c)) ----
  float sc = 0.f, sc2 = 0.f, ss = 0.f;
#pragma unroll
  for (int i = 0; i < EPT; ++i) {
    if ((mb >> i) & 1ull) {
      const float c = v[i] - mean1;
      v[i] = c;
      sc  += c;
      sc2 += c * c;
      ss  += (c > 0.f ? 1.f : 0.f) - (c < 0.f ? 1.f : 0.f);
    }
  }
  block_reduce3(sc, sc2, ss, red);
  const float scale1 = sqrtf(sc2 * inv_cnt) * SQRT_2_OVER_PI;
  // residual2 = c1 - scale1*sign(c1)  =>  sum(residual2) = sc - scale1*ss
  const float mean2  = (sc - scale1 * ss) * inv_cnt;

  // ---- order 2: c2 = c1 - scale1*s1 - mean2; remember s1 as bitmasks ----
  uint64_t nz1 = 0ull, ng1 = 0ull;
  float t2 = 0.f, d1 = 0.f, d2 = 0.f;
#pragma unroll
  for (int i = 0; i < EPT; ++i) {
    if ((mb >> i) & 1ull) {
      const float c1 = v[i];
      const float s1 = (c1 > 0.f ? 1.f : 0.f) - (c1 < 0.f ? 1.f : 0.f);
      nz1 |= (c1 != 0.f) ? (1ull << i) : 0ull;
      ng1 |= (c1 <  0.f) ? (1ull << i) : 0ull;
      const float c2 = c1 - scale1 * s1 - mean2;
      v[i] = c2;
      t2  += c2 * c2;
    }
  }
  block_reduce3(t2, d1, d2, red);
  const float scale2 = sqrtf(t2 * inv_cnt) * SQRT_2_OVER_PI;

  // ---- final: out = mask ? mean1 + scale1*s1 + mean2 + scale2*s2 : 0 ----
  float* __restrict__ orow = out + (size_t)row * COLS;
  const float base = mean1 + mean2;
#pragma unroll
  for (int i = 0; i < EPT; ++i) {
    const int idx = tid + i * TPB;
    float o = 0.f;
    if ((mb >> i) & 1ull) {
      const float s1 = ((nz1 >> i) & 1ull) ? (((ng1 >> i) & 1ull) ? -1.f : 1.f) : 0.f;
      const float c2 = v[i];
      const float s2 = (c2 > 0.f ? 1.f : 0.f) - (c2 < 0.f ? 1.f : 0.f);
      o = base + scale1 * s1 + scale2 * s2;
    }
    orow[idx] = o;   // coalesced b32 stores (wave32 * 4B = 128B bursts)
  }
}

extern "C" void kernel_launch(void* const* d_in, const int* in_sizes, int n_in,
                              void* d_out, int out_size, void* d_ws, size_t ws_size,
                              hipStream_t stream) {
  (void)n_in; (void)d_ws; (void)ws_size; (void)out_size;
  const float*         xp = (const float*)d_in[0];
  const unsigned char* mp = (const unsigned char*)d_in[1];  // jnp.bool_ -> 1 byte/elem
  float*               op = (float*)d_out;
  const int rows = in_sizes[0] / COLS;                      // 4096
  hipLaunchKernelGGL(binarize2_rowkernel, dim3(rows), dim3(TPB), 0, stream, xp, mp, op);
}